// AdaptedEntropyModel_31490700214747
// MI455X (gfx1250) — compile-verified
//
#include <hip/hip_runtime.h>
#include <cfloat>

// ---------------------------------------------------------------------------
// AdaptedEntropyModel for MI455X (gfx1250, wave32).
//
// Roofline: quantize pass moves 16 B/element * 2,359,296 elements = 37.7 MB
// -> ~1.6 us at 23.3 TB/s. Purely bandwidth-bound; WMMA has no role (no GEMM
// structure). CDNA5-specific paths used instead:
//   * global_load_async_to_lds_b64 + s_wait_asynccnt  (async table staging)
//   * non-temporal b128 loads/stores (streaming, keep 192MB L2 clean)
//   * global_prefetch_b8 via __builtin_prefetch
//   * per-lane ds_load binary search (wave32 LDS gather)
// ---------------------------------------------------------------------------

#define NCHAN   192
#define PMF_ML  64      // MAXLEN
#define LVALS   60      // number of unique_values
#define PREC_SCALE 65536.0f   // 1 << 16

// Native clang vector types (HIP's float4/int4 are structs and are rejected
// by __builtin_nontemporal_load/store).
typedef float v4f __attribute__((ext_vector_type(4)));
typedef int   v4i __attribute__((ext_vector_type(4)));

typedef float __attribute__((address_space(3))) *lds_fptr;

// Branchless lower-bound count over a 64-entry sorted LDS table padded with
// +FLT_MAX: returns #{j : mids[j] < v}  == jnp.searchsorted(mids, v, 'left').
__device__ __forceinline__ int lower_count64(const float *mids, float v) {
    int p = 0;
#pragma unroll
    for (int step = 32; step >= 1; step >>= 1) {
        p += (mids[p + step - 1] < v) ? step : 0;   // ds_load_b32 gather
    }
    return p;
}

__global__ __launch_bounds__(256)
void quantize_kernel(const float *__restrict__ in,
                     const float *__restrict__ mn,
                     const float *__restrict__ uv,
                     int   *__restrict__ sym_out,
                     float *__restrict__ deq_out,
                     long n4)
{
    __shared__ __align__(16) float lds_uv[64];
    __shared__ __align__(16) float lds_mids[64];

    const int tid = threadIdx.x;

    // ---- Stage unique_values (60 floats) into LDS with the CDNA5 async
    //      global->LDS copy path (tracked on ASYNCcnt, not LOADcnt). ----
    if (tid < (LVALS / 2)) {                 // 30 lanes x b64 = 60 floats
        lds_fptr dst = (lds_fptr)&lds_uv[2 * tid];
        unsigned long long src = (unsigned long long)(const void *)(uv + 2 * tid);
        asm volatile("global_load_async_to_lds_b64 %0, %1, off"
                     :: "v"(dst), "v"(src) : "memory");
    }
    asm volatile("s_wait_asynccnt 0x0" ::: "memory");
    __syncthreads();

    // Build midpoint table, padded to 64 with FLT_MAX so the 6-step branchless
    // search needs no bounds checks.
    if (tid < 64) {
        lds_mids[tid] = (tid < LVALS - 1)
                            ? 0.5f * (lds_uv[tid + 1] + lds_uv[tid])
                            : FLT_MAX;
    }
    __syncthreads();

    const v4f *in4 = (const v4f *)in;
    const v4f *mn4 = (const v4f *)mn;
    v4i *sym4 = (v4i *)sym_out;
    v4f *deq4 = (v4f *)deq_out;

    const long stride = (long)gridDim.x * (long)blockDim.x;
    for (long i = (long)blockIdx.x * blockDim.x + tid; i < n4; i += stride) {
        if (i + stride < n4) {               // global_prefetch_b8 (speculative)
            __builtin_prefetch(in4 + i + stride, 0, 0);
            __builtin_prefetch(mn4 + i + stride, 0, 0);
        }
        const v4f x = __builtin_nontemporal_load(in4 + i);
        const v4f m = __builtin_nontemporal_load(mn4 + i);

        const float c0 = x.x - m.x;
        const float c1 = x.y - m.y;
        const float c2 = x.z - m.z;
        const float c3 = x.w - m.w;

        const int s0 = lower_count64(lds_mids, c0);
        const int s1 = lower_count64(lds_mids, c1);
        const int s2 = lower_count64(lds_mids, c2);
        const int s3 = lower_count64(lds_mids, c3);

        v4i s;
        s.x = s0; s.y = s1; s.z = s2; s.w = s3;
        v4f d;
        d.x = lds_uv[s0] + m.x;
        d.y = lds_uv[s1] + m.y;
        d.z = lds_uv[s2] + m.z;
        d.w = lds_uv[s3] + m.w;

        __builtin_nontemporal_store(s, sym4 + i);
        __builtin_nontemporal_store(d, deq4 + i);
    }
}

// ---- CDF table: 192 channels x 66 int32. One thread per channel, two serial
//      65-step passes (compute total, then scaled cumsum). Latency-trivial. --
__global__ __launch_bounds__(NCHAN)
void cdf_kernel(const float *__restrict__ pmf,
                const float *__restrict__ tail_mass,
                const int   *__restrict__ pmf_length,
                int *__restrict__ cdf)
{
    const int c = threadIdx.x;
    if (c >= NCHAN) return;

    const float *row  = pmf + (long)c * PMF_ML;
    const int    plen = pmf_length[c];
    const float  tail = tail_mass[c];

    // Pass 1: total = sum of masked probs (same summation order as cumsum[-1]).
    float total = 0.0f;
    for (int i = 0; i <= PMF_ML; ++i) {
        float p = (i < plen && i < PMF_ML) ? row[i] : 0.0f;
        if (i == plen) p += tail;
        total += p;
    }

    int *out = cdf + (long)c * (PMF_ML + 2);
    out[0] = 0;

    // Pass 2: cdf_body[i] = round(cum/total * 2^16), round-half-even (rintf),
    // zeroed where position > plen+1.
    float cum = 0.0f;
    for (int i = 0; i <= PMF_ML; ++i) {
        float p = (i < plen && i < PMF_ML) ? row[i] : 0.0f;
        if (i == plen) p += tail;
        cum += p;
        const int val = (int)rintf(cum / total * PREC_SCALE);
        out[i + 1] = (i <= plen) ? val : 0;
    }
}

extern "C" void kernel_launch(void *const *d_in, const int *in_sizes, int n_in,
                              void *d_out, int out_size, void *d_ws, size_t ws_size,
                              hipStream_t stream)
{
    const float *in   = (const float *)d_in[0];   // inputs  (B,C,H,W) f32
    const float *mn   = (const float *)d_in[1];   // means   (B,C,H,W) f32
    const float *uv   = (const float *)d_in[2];   // unique_values (60,) f32
    const float *pmf  = (const float *)d_in[3];   // (192,64) f32
    const float *tail = (const float *)d_in[4];   // (192,1) f32
    const int   *plen = (const int *)d_in[5];     // (192,) i32

    const long N  = (long)in_sizes[0];            // 2,359,296 (divisible by 4)
    const long n4 = N / 4;

    int   *sym = (int *)d_out;                    // tuple order: sym | deq | cdf
    float *deq = (float *)((int *)d_out + N);
    int   *cdf = (int *)d_out + 2 * N;

    const int  block  = 256;                      // 8 wave32 waves / block
    const long blocks = (n4 + block - 1) / block; // 2304 blocks

    quantize_kernel<<<dim3((unsigned)blocks), dim3(block), 0, stream>>>(
        in, mn, uv, sym, deq, n4);
    cdf_kernel<<<dim3(1), dim3(NCHAN), 0, stream>>>(pmf, tail, plen, cdf);
}